// PrototypeLayer_32238024524557
// MI455X (gfx1250) — compile-verified
//
#include <hip/hip_runtime.h>
#include <hip/hip_bf16.h>
#include <math.h>

typedef __attribute__((ext_vector_type(2))) float v2f;
typedef __attribute__((ext_vector_type(8))) float v8f;

#define NSAMP 4096
#define INDIM 768
#define DDIM  256
#define PROTO 64
#define EPSV  1e-8f
#define TOK_ELEMS ((size_t)NSAMP * PROTO * DDIM)

// ---------------------------------------------------------------------------
// z = x @ W + b  via V_WMMA_F32_16X16X4_F32.
// Register-blocked: each wave32 computes a 16(M) x 64(N) tile with 4 v8f
// accumulators, so one A-load feeds 4 WMMAs (4x A reuse).
// A layout (ISA 7.12.2, 32-bit A 16x4): lane l<16 -> {A[m=l][k+0],A[m=l][k+1]},
// lane l+16 -> {A[m=l][k+2],A[m=l][k+3]}.  B mirrored with N across lanes.
// ---------------------------------------------------------------------------
__global__ __launch_bounds__(32) void gemm_z_kernel(const float* __restrict__ x,
                                                    const float* __restrict__ W,
                                                    const float* __restrict__ bias,
                                                    float* __restrict__ z) {
  const int tile = blockIdx.x;          // 256 M-tiles x 4 N-supertiles
  const int tm  = (tile >> 2) * 16;
  const int tn0 = (tile & 3) * 64;
  const int lane = threadIdx.x & 31;
  const int l15  = lane & 15;
  const int lh   = lane >> 4;
  v8f acc0 = {}, acc1 = {}, acc2 = {}, acc3 = {};
  for (int k = 0; k < INDIM; k += 4) {
    const int ka = k + lh * 2;
    const v2f a = *(const v2f*)(x + (size_t)(tm + l15) * INDIM + ka);
    const float* w0 = W + (size_t)ka * DDIM + tn0 + l15;
    const float* w1 = w0 + DDIM;
    v2f b0, b1, b2, b3;
    b0.x = w0[0];  b0.y = w1[0];
    b1.x = w0[16]; b1.y = w1[16];
    b2.x = w0[32]; b2.y = w1[32];
    b3.x = w0[48]; b3.y = w1[48];
    acc0 = __builtin_amdgcn_wmma_f32_16x16x4_f32(false, a, false, b0, (short)0, acc0, false, false);
    acc1 = __builtin_amdgcn_wmma_f32_16x16x4_f32(false, a, false, b1, (short)0, acc1, false, false);
    acc2 = __builtin_amdgcn_wmma_f32_16x16x4_f32(false, a, false, b2, (short)0, acc2, false, false);
    acc3 = __builtin_amdgcn_wmma_f32_16x16x4_f32(false, a, false, b3, (short)0, acc3, false, false);
  }
  v8f accs[4] = {acc0, acc1, acc2, acc3};
#pragma unroll
  for (int j = 0; j < 4; ++j) {
    const int col = tn0 + j * 16 + l15;
    const float bv = bias[col];
#pragma unroll
    for (int i = 0; i < 8; ++i) {
      const int row = tm + i + lh * 8;  // C layout: vgpr i -> M=i (lanes<16) / M=i+8
      z[(size_t)row * DDIM + col] = accs[j][i] + bv;
    }
  }
}

// ---------------------------------------------------------------------------
// G = z @ prototypes^T  (Gram matrix). One wave covers a 16(M) x 64(P) tile,
// i.e. ALL prototypes, so each z row is read exactly once.
// ---------------------------------------------------------------------------
__global__ __launch_bounds__(32) void gemm_g_kernel(const float* __restrict__ z,
                                                    const float* __restrict__ proto,
                                                    float* __restrict__ G) {
  const int tm = blockIdx.x * 16;       // 256 M-tiles
  const int lane = threadIdx.x & 31;
  const int l15  = lane & 15;
  const int lh   = lane >> 4;
  v8f acc0 = {}, acc1 = {}, acc2 = {}, acc3 = {};
  for (int k = 0; k < DDIM; k += 4) {
    const int ka = k + lh * 2;
    const v2f a = *(const v2f*)(z + (size_t)(tm + l15) * DDIM + ka);
    // B[k][n] = proto[n][k] -> consecutive pair in proto row
    const v2f b0 = *(const v2f*)(proto + (size_t)(l15 +  0) * DDIM + ka);
    const v2f b1 = *(const v2f*)(proto + (size_t)(l15 + 16) * DDIM + ka);
    const v2f b2 = *(const v2f*)(proto + (size_t)(l15 + 32) * DDIM + ka);
    const v2f b3 = *(const v2f*)(proto + (size_t)(l15 + 48) * DDIM + ka);
    acc0 = __builtin_amdgcn_wmma_f32_16x16x4_f32(false, a, false, b0, (short)0, acc0, false, false);
    acc1 = __builtin_amdgcn_wmma_f32_16x16x4_f32(false, a, false, b1, (short)0, acc1, false, false);
    acc2 = __builtin_amdgcn_wmma_f32_16x16x4_f32(false, a, false, b2, (short)0, acc2, false, false);
    acc3 = __builtin_amdgcn_wmma_f32_16x16x4_f32(false, a, false, b3, (short)0, acc3, false, false);
  }
  v8f accs[4] = {acc0, acc1, acc2, acc3};
#pragma unroll
  for (int j = 0; j < 4; ++j) {
    const int col = j * 16 + l15;
#pragma unroll
    for (int i = 0; i < 8; ++i) {
      const int row = tm + i + lh * 8;
      G[(size_t)row * PROTO + col] = accs[j][i];
    }
  }
}

// ---------------------------------------------------------------------------
// Squared L2 norms of z rows (blocks 0..4095) and prototype rows (4096..4159).
// ---------------------------------------------------------------------------
__global__ void norms_kernel(const float* __restrict__ z,
                             const float* __restrict__ proto,
                             float* __restrict__ zn2, float* __restrict__ pn2) {
  __shared__ float s[256];
  const int bid = blockIdx.x, t = threadIdx.x;
  const float* src;
  float* dst;
  int row;
  if (bid < NSAMP) { src = z; row = bid; dst = zn2 + bid; }
  else             { src = proto; row = bid - NSAMP; dst = pn2 + row; }
  const float v = src[(size_t)row * DDIM + t];
  s[t] = v * v;
  __syncthreads();
  for (int o = 128; o > 0; o >>= 1) { if (t < o) s[t] += s[t + o]; __syncthreads(); }
  if (t == 0) *dst = s[0];
}

// ---------------------------------------------------------------------------
// Per-sample stats: min dist over P, count(cos>0.5), class-wise max cos.
// One wave per sample; lane handles p=lane (pos half) and p=lane+32 (neg half).
// ---------------------------------------------------------------------------
__global__ __launch_bounds__(32) void rowstats_kernel(const float* __restrict__ G,
                                                      const float* __restrict__ zn2a,
                                                      const float* __restrict__ pn2a,
                                                      const int* __restrict__ labels,
                                                      float* __restrict__ minDist,
                                                      float* __restrict__ numFlag,
                                                      float* __restrict__ clusterC) {
  const int n = blockIdx.x;
  const int lane = threadIdx.x & 31;
  const float zn2 = zn2a[n];
  const float znrm = fmaxf(sqrtf(zn2), EPSV);
  const float g0 = G[(size_t)n * PROTO + lane];
  const float g1 = G[(size_t)n * PROTO + lane + 32];
  const float q0 = pn2a[lane], q1 = pn2a[lane + 32];
  const float c0 = g0 / (znrm * fmaxf(sqrtf(q0), EPSV));
  const float c1 = g1 / (znrm * fmaxf(sqrtf(q1), EPSV));
  const float d0 = sqrtf(fmaxf(zn2 + q0 - 2.0f * g0, 0.0f));
  const float d1 = sqrtf(fmaxf(zn2 + q1 - 2.0f * g1, 0.0f));
  float mind = fminf(d0, d1);
  float cnt  = (c0 > 0.5f ? 1.0f : 0.0f) + (c1 > 0.5f ? 1.0f : 0.0f);
  float mp = c0, mn = c1;
  for (int m = 16; m >= 1; m >>= 1) {
    mind = fminf(mind, __shfl_xor(mind, m, 32));
    cnt += __shfl_xor(cnt, m, 32);
    mp = fmaxf(mp, __shfl_xor(mp, m, 32));
    mn = fmaxf(mn, __shfl_xor(mn, m, 32));
  }
  if (lane == 0) {
    minDist[n]  = mind;
    numFlag[n]  = (cnt > 16.0f) ? 1.0f : 0.0f;          // num_sim > P/4
    clusterC[n] = 1.0f - (labels[n] == 1 ? mp : mn);
  }
}

// ---------------------------------------------------------------------------
// Per-prototype min distance over all samples (for l_zp).
// ---------------------------------------------------------------------------
__global__ void colmin_kernel(const float* __restrict__ G,
                              const float* __restrict__ zn2a,
                              const float* __restrict__ pn2a,
                              float* __restrict__ colMin) {
  __shared__ float s[256];
  const int p = blockIdx.x, t = threadIdx.x;
  const float q = pn2a[p];
  float m = 3.4e38f;
  for (int n = t; n < NSAMP; n += 256) {
    const float g = G[(size_t)n * PROTO + p];
    m = fminf(m, sqrtf(fmaxf(zn2a[n] + q - 2.0f * g, 0.0f)));
  }
  s[t] = m;
  __syncthreads();
  for (int o = 128; o > 0; o >>= 1) { if (t < o) s[t] = fminf(s[t], s[t + o]); __syncthreads(); }
  if (t == 0) colMin[p] = s[0];
}

// ---------------------------------------------------------------------------
// Mean off-diagonal pairwise prototype distance (l_pp). Single block.
// ---------------------------------------------------------------------------
__global__ void pp_kernel(const float* __restrict__ proto,
                          const float* __restrict__ pn2a,
                          float* __restrict__ ppOut) {
  __shared__ float s[256];
  const int t = threadIdx.x;
  float acc = 0.0f;
  for (int idx = t; idx < PROTO * PROTO; idx += 256) {
    const int i = idx >> 6, j = idx & 63;
    if (i == j) continue;
    const float4* a = (const float4*)(proto + (size_t)i * DDIM);
    const float4* b = (const float4*)(proto + (size_t)j * DDIM);
    float dot = 0.0f;
    for (int d = 0; d < DDIM / 4; ++d) {
      const float4 av = a[d], bv = b[d];
      dot += av.x * bv.x + av.y * bv.y + av.z * bv.z + av.w * bv.w;
    }
    acc += sqrtf(fmaxf(pn2a[i] + pn2a[j] - 2.0f * dot, 0.0f));
  }
  s[t] = acc;
  __syncthreads();
  for (int o = 128; o > 0; o >>= 1) { if (t < o) s[t] += s[t + o]; __syncthreads(); }
  if (t == 0) *ppOut = s[0] / (float)(PROTO * (PROTO - 1));
}

// ---------------------------------------------------------------------------
// token_sequences[n][p][d] = cos[n][p] * proto[p][d]  — the 268 MB write.
// Fully coalesced float4 stores; cos recomputed from G + cached norms.
// ---------------------------------------------------------------------------
__global__ void token_kernel(const float* __restrict__ G,
                             const float* __restrict__ zn2a,
                             const float* __restrict__ pn2a,
                             const float* __restrict__ proto,
                             float* __restrict__ out) {
  __shared__ float cosRow[PROTO];
  const int n = blockIdx.x, t = threadIdx.x;
  if (t < PROTO) {
    const float znrm = fmaxf(sqrtf(zn2a[n]), EPSV);
    cosRow[t] = G[(size_t)n * PROTO + t] / (znrm * fmaxf(sqrtf(pn2a[t]), EPSV));
  }
  __syncthreads();
  const int sub = t >> 6;           // 0..3 : which prototype of this group of 4
  const int d4  = (t & 63) * 4;     // float4 column
  float* base = out + (size_t)n * PROTO * DDIM;
#pragma unroll
  for (int it = 0; it < 16; ++it) {
    const int p = it * 4 + sub;
    const float c = cosRow[p];
    const float4 pv = *(const float4*)(proto + (size_t)p * DDIM + d4);
    float4 o;
    o.x = c * pv.x; o.y = c * pv.y; o.z = c * pv.z; o.w = c * pv.w;
    *(float4*)(base + (size_t)p * DDIM + d4) = o;
  }
}

// ---------------------------------------------------------------------------
// Deterministic final combine -> scalar loss.
// ---------------------------------------------------------------------------
__global__ void finalize_kernel(const float* __restrict__ minDist,
                                const float* __restrict__ numFlag,
                                const float* __restrict__ clusterC,
                                const float* __restrict__ colMin,
                                const float* __restrict__ ppOut,
                                float* __restrict__ lossOut) {
  __shared__ float s1[256], s2[256], s3[256], s4[256];
  const int t = threadIdx.x;
  float a = 0.0f, b = 0.0f, c = 0.0f;
  for (int n = t; n < NSAMP; n += 256) { a += minDist[n]; b += numFlag[n]; c += clusterC[n]; }
  s1[t] = a; s2[t] = b; s3[t] = c; s4[t] = (t < PROTO) ? colMin[t] : 0.0f;
  __syncthreads();
  for (int o = 128; o > 0; o >>= 1) {
    if (t < o) { s1[t] += s1[t + o]; s2[t] += s2[t + o]; s3[t] += s3[t + o]; s4[t] += s4[t + o]; }
    __syncthreads();
  }
  if (t == 0) {
    const float l_pz      = s1[0] / (float)NSAMP;
    const float l_num     = s2[0];
    const float l_cluster = s3[0] / (float)NSAMP;
    const float l_zp      = s4[0] / (float)PROTO;
    // W_SEP also multiplies cluster loss (replicated upstream bug)
    *lossOut = l_zp + l_pz + *ppOut + l_num + 2.0f * l_cluster;
  }
}

extern "C" void kernel_launch(void* const* d_in, const int* in_sizes, int n_in,
                              void* d_out, int out_size, void* d_ws, size_t ws_size,
                              hipStream_t stream) {
  const float* x      = (const float*)d_in[0];   // [4096,768]
  const int*   labels = (const int*)d_in[1];     // [4096]
  const float* W      = (const float*)d_in[2];   // [768,256]
  const float* b      = (const float*)d_in[3];   // [256]
  const float* proto  = (const float*)d_in[4];   // [64,256]
  float* out  = (float*)d_out;                   // tokens then loss
  float* loss = out + TOK_ELEMS;

  // Workspace layout (floats)
  float* z        = (float*)d_ws;                  // 4096*256
  float* G        = z + (size_t)NSAMP * DDIM;      // 4096*64
  float* zn2      = G + (size_t)NSAMP * PROTO;     // 4096
  float* pn2      = zn2 + NSAMP;                   // 64
  float* minDist  = pn2 + PROTO;                   // 4096
  float* numFlag  = minDist + NSAMP;               // 4096
  float* clusterC = numFlag + NSAMP;               // 4096
  float* colMin   = clusterC + NSAMP;              // 64
  float* ppOut    = colMin + PROTO;                // 1

  gemm_z_kernel<<<(NSAMP / 16) * (DDIM / 64), 32, 0, stream>>>(x, W, b, z);
  norms_kernel<<<NSAMP + PROTO, 256, 0, stream>>>(z, proto, zn2, pn2);
  gemm_g_kernel<<<NSAMP / 16, 32, 0, stream>>>(z, proto, G);
  rowstats_kernel<<<NSAMP, 32, 0, stream>>>(G, zn2, pn2, labels, minDist, numFlag, clusterC);
  colmin_kernel<<<PROTO, 256, 0, stream>>>(G, zn2, pn2, colMin);
  pp_kernel<<<1, 256, 0, stream>>>(proto, pn2, ppOut);
  token_kernel<<<NSAMP, 256, 0, stream>>>(G, zn2, pn2, proto, out);
  finalize_kernel<<<1, 256, 0, stream>>>(minDist, numFlag, clusterC, colMin, ppOut, loss);
}